// GraphProjection_28750511079682
// MI455X (gfx1250) — compile-verified
//
#include <hip/hip_runtime.h>
#include <hip/hip_bf16.h>
#include <math.h>

typedef __attribute__((ext_vector_type(2))) float v2f;
typedef __attribute__((ext_vector_type(4))) float v4f;
typedef __attribute__((ext_vector_type(8))) float v8f;

#define PI_F 3.14159265358979323846f

// ---------------------------------------------------------------------------
// Setup: fold camera trig + inv(c0^T) into one 4x16 WMMA B-fragment.
// B[k][3v+j] = R_v[k][j] (k<3),  B[3][3v+j] = t_v[j];  columns 9..15 zero.
// pc_v[j] = sum_k coord[k]*R_v[k][j] + t_v[j]  ==  D = A(16x4) x B(4x16)
// ---------------------------------------------------------------------------
__global__ void gp_setup_kernel(const float* __restrict__ cam,
                                float* __restrict__ B) {
    if (threadIdx.x != 0 || blockIdx.x != 0) return;

    float c[3][3][3];   // c[view][row][col]
    float o[3][3];      // o[view][xyz]
    for (int i = 0; i < 3; ++i) {
        float az = cam[i * 5 + 0], el = cam[i * 5 + 1], dist = cam[i * 5 + 3];
        float th = az * PI_F / 180.0f;
        float er = el * PI_F / 180.0f;
        float camy = dist * sinf(er);
        float lens = dist * cosf(er);
        float camx = lens * cosf(th);
        float camz = lens * sinf(th);
        float Zv[3] = {camx, camy, camz};
        float Yv[3] = {camy * cosf(th + PI_F), lens, camy * sinf(th + PI_F)};
        float Xv[3] = {Yv[1] * Zv[2] - Yv[2] * Zv[1],
                       Yv[2] * Zv[0] - Yv[0] * Zv[2],
                       Yv[0] * Zv[1] - Yv[1] * Zv[0]};
        float* rows[3] = {Xv, Yv, Zv};
        for (int r = 0; r < 3; ++r) {
            float* v = rows[r];
            float n = sqrtf(v[0] * v[0] + v[1] * v[1] + v[2] * v[2]);
            c[i][r][0] = v[0] / n;
            c[i][r][1] = v[1] / n;
            c[i][r][2] = v[2] / n;
        }
        o[i][0] = Zv[0]; o[i][1] = Zv[1]; o[i][2] = Zv[2];
    }

    // A0 = c0^T ; inv via adjugate/det (c0 is orthonormal, but match linalg.inv)
    float A[3][3];
    for (int k = 0; k < 3; ++k)
        for (int m = 0; m < 3; ++m)
            A[k][m] = c[0][m][k];
    float det = A[0][0] * (A[1][1] * A[2][2] - A[1][2] * A[2][1])
              - A[0][1] * (A[1][0] * A[2][2] - A[1][2] * A[2][0])
              + A[0][2] * (A[1][0] * A[2][1] - A[1][1] * A[2][0]);
    float inv[3][3];
    inv[0][0] =  (A[1][1] * A[2][2] - A[1][2] * A[2][1]) / det;
    inv[0][1] = -(A[0][1] * A[2][2] - A[0][2] * A[2][1]) / det;
    inv[0][2] =  (A[0][1] * A[1][2] - A[0][2] * A[1][1]) / det;
    inv[1][0] = -(A[1][0] * A[2][2] - A[1][2] * A[2][0]) / det;
    inv[1][1] =  (A[0][0] * A[2][2] - A[0][2] * A[2][0]) / det;
    inv[1][2] = -(A[0][0] * A[1][2] - A[0][2] * A[1][0]) / det;
    inv[2][0] =  (A[1][0] * A[2][1] - A[1][1] * A[2][0]) / det;
    inv[2][1] = -(A[0][0] * A[2][1] - A[0][1] * A[2][0]) / det;
    inv[2][2] =  (A[0][0] * A[1][1] - A[0][1] * A[1][0]) / det;

    for (int n = 0; n < 64; ++n) B[n] = 0.0f;   // zero-pad cols 9..15

    for (int v = 0; v < 3; ++v) {
        for (int j = 0; j < 3; ++j) {
            for (int k = 0; k < 3; ++k) {
                float r = inv[k][0] * c[v][j][0]
                        + inv[k][1] * c[v][j][1]
                        + inv[k][2] * c[v][j][2];
                B[k * 16 + 3 * v + j] = r;
            }
            float t = (o[0][0] - o[v][0]) * c[v][j][0]
                    + (o[0][1] - o[v][1]) * c[v][j][1]
                    + (o[0][2] - o[v][2]) * c[v][j][2];
            B[3 * 16 + 3 * v + j] = t;
        }
    }
}

// Process one 4-channel group: 3x b128 gather, per-component max/mean/std,
// 12x b32 stores (wave-coalesced; row layout forbids aligned b128 stores).
__device__ __forceinline__ void gp_group(const v4f* __restrict__ b0,
                                         const v4f* __restrict__ b1,
                                         const v4f* __restrict__ b2,
                                         int g, int cstart,
                                         float* __restrict__ orow) {
    v4f x = b0[g];
    v4f y = b1[g];
    v4f z = b2[g];
    int c0 = cstart + g * 4;
    #pragma unroll
    for (int j = 0; j < 4; ++j) {
        float v0 = x[j], v1 = y[j], v2 = z[j];
        float mx = fmaxf(v0, fmaxf(v1, v2));
        float mn = (v0 + v1 + v2) * (1.0f / 3.0f);
        float d0 = v0 - mn, d1 = v1 - mn, d2 = v2 - mn;
        float sd = sqrtf((d0 * d0 + d1 * d1 + d2 * d2) * (1.0f / 3.0f));
        orow[3 + c0 + j]    = mx;
        orow[963 + c0 + j]  = mn;
        orow[1923 + c0 + j] = sd;
    }
}

// ---------------------------------------------------------------------------
// Main kernel: one wave = 16 points. WMMA does the 3-view affine transform,
// then 32 lanes cooperatively gather 960 channels/point (b128 gathers).
// ---------------------------------------------------------------------------
__global__ __launch_bounds__(256)
void gp_main_kernel(const float* __restrict__ coord,
                    const float* __restrict__ Bws,
                    const float* __restrict__ f1,
                    const float* __restrict__ f2,
                    const float* __restrict__ f3,
                    const float* __restrict__ f4,
                    float* __restrict__ out,
                    int npoints) {
    __shared__ float lds[8][256];   // 8 waves x (16 points x 16 cols)

    const int lane = threadIdx.x & 31;
    const int wv   = threadIdx.x >> 5;
    const int base = (blockIdx.x * 8 + wv) * 16;
    const int pm   = lane & 15;
    const int hi   = lane >> 4;     // 0: lanes 0-15, 1: lanes 16-31

    // --- A fragment: 16x4 f32, lane layout per ISA (branchless; EXEC all-1) ---
    int praw = base + pm;
    int pidx = (praw < npoints) ? praw : (npoints - 1);
    const float* cp = coord + 3ll * pidx;
    float cx = cp[0], cy = cp[1], cz = cp[2];
    v2f a;
    a.x = hi ? cz   : cx;
    a.y = hi ? 1.0f : cy;

    // --- B fragment: 4x16 f32. VGPR0 = rows K={0,2}, VGPR1 = rows K={1,3} ---
    v2f b;
    {
        int col = lane & 15;
        int r0 = hi * 2;
        b.x = Bws[r0 * 16 + col];
        b.y = Bws[(r0 + 1) * 16 + col];
    }

    v8f cacc = {};
    // D = A x B + 0  -> v_wmma_f32_16x16x4_f32
    v8f d = __builtin_amdgcn_wmma_f32_16x16x4_f32(
        false, a, false, b, (short)0, cacc, false, false);

    // --- spill D to LDS: VGPR r holds M=r (lanes 0-15) / M=r+8 (lanes 16-31) ---
    float* slot = lds[wv];
    {
        int M0 = hi * 8;
        int Nc = lane & 15;
        #pragma unroll
        for (int r = 0; r < 8; ++r)
            slot[(M0 + r) * 16 + Nc] = d[r];
    }
    __syncthreads();

    if (base >= npoints) return;    // wave-uniform

    // --- coord passthrough: each lane holds its point's xyz already ---
    if (praw < npoints) {
        float* prow = out + (long long)praw * 2883;
        if (hi == 0) { prow[0] = cx; prow[1] = cy; }
        else         { prow[2] = cz; }
    }

    const float* fptr[4] = {f1, f2, f3, f4};
    const int dims[4] = {56, 28, 14, 7};
    const int chs[4]  = {64, 128, 256, 512};

    for (int q = 0; q < 16; ++q) {
        const int p = base + q;
        if (p >= npoints) break;    // wave-uniform
        float* orow = out + (long long)p * 2883;

        // per-view, per-level gather base pointers (lane-uniform, 16B aligned)
        const v4f* vb[3][4];
        #pragma unroll
        for (int v = 0; v < 3; ++v) {
            float X  = slot[q * 16 + v * 3 + 0];
            float Y  = slot[q * 16 + v * 3 + 1];
            float Zc = slot[q * 16 + v * 3 + 2];
            float h = (248.0f * (-Y)) / (-Zc) + 112.0f;
            float w = (248.0f * ( X)) / (-Zc) + 112.0f;
            h = fminf(fmaxf(h, 0.0f), 223.0f);
            w = fminf(fmaxf(w, 0.0f), 223.0f);
            float nf = (float)v;
            #pragma unroll
            for (int l = 0; l < 4; ++l) {
                float sc = 224.0f / (float)dims[l];
                int i0 = (int)(nf / sc);     // faithfully == 0, as in reference
                int ih = (int)(h / sc);
                int iw = (int)(w / sc);
                vb[v][l] = (const v4f*)(fptr[l]
                         + (long long)((i0 * dims[l] + ih) * dims[l] + iw) * chs[l]);
            }
        }

        // f1: 64 ch = 16 groups -> half wave, 1 iter
        if (lane < 16)
            gp_group(vb[0][0], vb[1][0], vb[2][0], lane, 0, orow);
        // f2: 128 ch = 32 groups -> 1 iter
        gp_group(vb[0][1], vb[1][1], vb[2][1], lane, 64, orow);
        // f3: 256 ch = 64 groups -> 2 iters
        #pragma unroll
        for (int it = 0; it < 2; ++it)
            gp_group(vb[0][2], vb[1][2], vb[2][2], lane + 32 * it, 192, orow);
        // f4: 512 ch = 128 groups -> 4 iters
        #pragma unroll
        for (int it = 0; it < 4; ++it)
            gp_group(vb[0][3], vb[1][3], vb[2][3], lane + 32 * it, 448, orow);
    }
}

extern "C" void kernel_launch(void* const* d_in, const int* in_sizes, int n_in,
                              void* d_out, int out_size, void* d_ws, size_t ws_size,
                              hipStream_t stream) {
    const float* coord   = (const float*)d_in[0];
    const float* cameras = (const float*)d_in[1];
    const float* f1      = (const float*)d_in[2];
    const float* f2      = (const float*)d_in[3];
    const float* f3      = (const float*)d_in[4];
    const float* f4      = (const float*)d_in[5];
    float* out = (float*)d_out;
    float* Bws = (float*)d_ws;     // 64 floats: 4x16 WMMA B fragment

    gp_setup_kernel<<<1, 32, 0, stream>>>(cameras, Bws);

    int npoints = in_sizes[0] / 3;                 // 80000
    int blocks  = (npoints + 127) / 128;           // 8 waves x 16 points each
    gp_main_kernel<<<blocks, 256, 0, stream>>>(coord, Bws, f1, f2, f3, f4,
                                               out, npoints);
}